// LocalGraphSampler_88450556494513
// MI455X (gfx1250) — compile-verified
//
#include <hip/hip_runtime.h>
#include <hip/hip_bf16.h>

// ---------------------------------------------------------------------------
// LocalGraphSampler for MI455X (gfx1250, wave32).
//   inputs : d_in[0] all_embeddings f32 [N=50000,64]
//            d_in[1] edge_index     i32 [2,E=800000]  (JAX x64-off -> int32)
//            d_in[2] edge_weight    f32 [E]           (unused, only shape)
//            d_in[3] n_seeds        i32 [1]           (2048)
//   output : d_out = scores f32 [N] ++ seeds (int32 bits) [n_seeds]
// ---------------------------------------------------------------------------

typedef __attribute__((ext_vector_type(2))) float v2f;
typedef __attribute__((ext_vector_type(8))) float v8f;

#ifndef __has_builtin
#define __has_builtin(x) 0
#endif
#if __has_builtin(__builtin_amdgcn_wmma_f32_16x16x4_f32)
#define HAVE_WMMA_F32X4 1
#else
#define HAVE_WMMA_F32X4 0
#endif

// ------------------------------ utils --------------------------------------

__device__ __forceinline__ unsigned rotl32(unsigned x, int n) {
  return (x << n) | (x >> (32 - n));
}

// JAX threefry2x32 with key = jax.random.key(1) -> (k0,k1) = (0,1)
__device__ __forceinline__ void threefry2x32_key1(unsigned c0, unsigned c1,
                                                  unsigned& o0, unsigned& o1) {
  const unsigned k0 = 0u, k1 = 1u;
  const unsigned k2 = 0x1BD11BDAu ^ k0 ^ k1;
  unsigned x0 = c0 + k0, x1 = c1 + k1;
#define TF_R(r) { x0 += x1; x1 = rotl32(x1, r); x1 ^= x0; }
  TF_R(13) TF_R(15) TF_R(26) TF_R(6)   x0 += k1; x1 += k2 + 1u;
  TF_R(17) TF_R(29) TF_R(16) TF_R(24)  x0 += k2; x1 += k0 + 2u;
  TF_R(13) TF_R(15) TF_R(26) TF_R(6)   x0 += k0; x1 += k1 + 3u;
  TF_R(17) TF_R(29) TF_R(16) TF_R(24)  x0 += k1; x1 += k2 + 4u;
  TF_R(13) TF_R(15) TF_R(26) TF_R(6)   x0 += k2; x1 += k0 + 5u;
#undef TF_R
  o0 = x0; o1 = x1;
}

// score epilogue shared by WMMA tile path and scalar tail path
__device__ __forceinline__ void finish_score(int node, int N, float dne,
                                             float dnn, float dee, float dg,
                                             float snumv, float* outScores,
                                             unsigned* keys) {
  // subgraph = (first+second)/(deg + second_num + 1e-8); numer = acc2 - deg*emb
  // cosine(normalize(numer/denom), normalize(emb)) = sign(denom)*dne/(|n||e|)
  float denom = snumv - dg + 1e-8f;
  float inv = rsqrtf(fmaxf(dnn, 1e-24f)) * rsqrtf(fmaxf(dee, 1e-24f));
  float cosv = dne * inv;
  if (denom < 0.0f) cosv = -cosv;
  float sig = 1.0f / (1.0f + __expf(-cosv));
  float ls = __logf(sig);
  // JAX uniform(key(1), (N,)): counters (i, i+half), bits>>9|0x3f800000 - 1
  unsigned half = ((unsigned)N + 1u) >> 1;
  unsigned i = (unsigned)node;
  unsigned c0 = (i < half) ? i : (i - half);
  unsigned c1 = (i < half) ? (i + half) : i;
  unsigned o0, o1;
  threefry2x32_key1(c0, c1, o0, o1);
  unsigned bits = (i < half) ? o0 : o1;
  float U = __uint_as_float((bits >> 9) | 0x3f800000u) - 1.0f;
  float sc = ls + (-__logf(-__logf(U)));
  outScores[node] = sc;
  unsigned ub = __float_as_uint(sc);
  keys[node] = (ub & 0x80000000u) ? ~ub : (ub | 0x80000000u);  // order-preserving
}

// ------------------------------ kernels ------------------------------------

__global__ void zero_kernel(unsigned* __restrict__ p, size_t n) {
  size_t i = (size_t)blockIdx.x * blockDim.x + threadIdx.x;
  size_t stride = (size_t)gridDim.x * blockDim.x;
  for (; i < n; i += stride) p[i] = 0u;
}

__global__ void init_state_kernel(unsigned* __restrict__ state,
                                  const int* __restrict__ nseeds) {
  if (threadIdx.x == 0) {
    int k = nseeds[0];
    if (k < 1) k = 1;
    if (k > 2048) k = 2048;
    state[0] = 0u;          // prefix
    state[1] = (unsigned)k; // remaining k
    state[2] = 0u;          // cntG
    state[3] = 0u;          // cntE
  }
}

__global__ void degree_kernel(const int* __restrict__ ei,
                              float* __restrict__ deg, int E) {
  int e = blockIdx.x * blockDim.x + threadIdx.x;
  if (e < E) atomicAdd(&deg[ei[e]], 1.0f);
}

// wave-per-edge SpMM: out[src] += x[dst]; 2 f32 components per lane
__global__ void spmm_edge_kernel(const int* __restrict__ ei,
                                 const float* __restrict__ x,
                                 float* __restrict__ acc, int E) {
  long long gid = (long long)blockIdx.x * blockDim.x + threadIdx.x;
  int e = (int)(gid >> 5);
  int lane = (int)(gid & 31);
  if (e >= E) return;
  int s = ei[e];
  int d = ei[(long long)E + e];
  const float2 v = *(const float2*)(x + (size_t)d * 64 + lane * 2);
  float* dst = acc + (size_t)s * 64 + lane * 2;
  atomicAdd(dst, v.x);
  atomicAdd(dst + 1, v.y);
}

__global__ void sub_kernel(float* __restrict__ acc,
                           const float* __restrict__ emb, long long n) {
  long long i = (long long)blockIdx.x * blockDim.x + threadIdx.x;
  if (i < n) acc[i] -= emb[i];
}

__global__ void spmm_num_kernel(const int* __restrict__ ei,
                                const float* __restrict__ deg,
                                float* __restrict__ snum, int E) {
  int e = blockIdx.x * blockDim.x + threadIdx.x;
  if (e < E) atomicAdd(&snum[ei[e]], deg[ei[(long long)E + e]]);
}

// 16 nodes per wave: three 16x16 Gram matrices via V_WMMA_F32_16X16X4_F32,
// diagonals give dot(numer,emb), |numer|^2, |emb|^2 in full f32.
__global__ void __launch_bounds__(256) score_tile_kernel(
    const float* __restrict__ emb, const float* __restrict__ acc2,
    const float* __restrict__ deg, const float* __restrict__ snum,
    float* __restrict__ outScores, unsigned* __restrict__ keys, int N) {
  const int lane = threadIdx.x & 31;
  const int tile = blockIdx.x * (blockDim.x >> 5) + (threadIdx.x >> 5);
  const int ntiles = N >> 4;
  if (tile >= ntiles) return;  // wave-uniform: EXEC all-ones inside

  const int m = lane & 15;
  const int node = tile * 16 + m;
  const int kOff = (lane >> 4) << 1;  // lanes 0-15: K 0,1 ; lanes 16-31: K 2,3
  const float dg = deg[node];
  const float* embRow = emb + (size_t)node * 64;
  const float* a2Row = acc2 + (size_t)node * 64;

#if HAVE_WMMA_F32X4
  v8f cne = {}, cnn = {}, cee = {};
#pragma unroll
  for (int kc = 0; kc < 16; ++kc) {
    const int k0 = kc * 4 + kOff;
    float2 e2 = *(const float2*)(embRow + k0);
    float2 a2 = *(const float2*)(a2Row + k0);
    v2f a, b;
    a.x = a2.x - dg * e2.x;  // numer = acc2 - deg*emb
    a.y = a2.y - dg * e2.y;
    b.x = e2.x;
    b.y = e2.y;
    cne = __builtin_amdgcn_wmma_f32_16x16x4_f32(false, a, false, b, (short)0,
                                                cne, false, false);
    cnn = __builtin_amdgcn_wmma_f32_16x16x4_f32(false, a, false, a, (short)0,
                                                cnn, false, false);
    cee = __builtin_amdgcn_wmma_f32_16x16x4_f32(false, b, false, b, (short)0,
                                                cee, false, false);
  }
  // diag[m]: VGPR r lane r (m=r<8) / VGPR r lane r+24 (m=r+8)
  float dne = 0.f, dnn = 0.f, dee = 0.f;
#pragma unroll
  for (int r = 0; r < 8; ++r) {
    float a0 = __shfl(cne[r], r, 32), a1 = __shfl(cne[r], r + 24, 32);
    float b0 = __shfl(cnn[r], r, 32), b1 = __shfl(cnn[r], r + 24, 32);
    float c0 = __shfl(cee[r], r, 32), c1 = __shfl(cee[r], r + 24, 32);
    if (lane == r)     { dne = a0; dnn = b0; dee = c0; }
    if (lane == r + 8) { dne = a1; dnn = b1; dee = c1; }
  }
#else
  // VALU fallback: per-lane partial sums, combine lane m with lane m+16
  float pne = 0.f, pnn = 0.f, pee = 0.f;
#pragma unroll
  for (int kc = 0; kc < 16; ++kc) {
    const int k0 = kc * 4 + kOff;
    float2 e2 = *(const float2*)(embRow + k0);
    float2 a2 = *(const float2*)(a2Row + k0);
    float ax = a2.x - dg * e2.x, ay = a2.y - dg * e2.y;
    pne += ax * e2.x + ay * e2.y;
    pnn += ax * ax + ay * ay;
    pee += e2.x * e2.x + e2.y * e2.y;
  }
  float dne = pne + __shfl(pne, m + 16, 32);
  float dnn = pnn + __shfl(pnn, m + 16, 32);
  float dee = pee + __shfl(pee, m + 16, 32);
#endif

  if (lane < 16) finish_score(node, N, dne, dnn, dee, dg, snum[node],
                              outScores, keys);
}

// scalar tail for N not divisible by 16 (N=50000 -> unused, kept for safety)
__global__ void score_tail_kernel(const float* __restrict__ emb,
                                  const float* __restrict__ acc2,
                                  const float* __restrict__ deg,
                                  const float* __restrict__ snum,
                                  float* __restrict__ outScores,
                                  unsigned* __restrict__ keys, int N,
                                  int start) {
  int node = start + blockIdx.x * blockDim.x + threadIdx.x;
  if (node >= N) return;
  float dg = deg[node];
  float dne = 0.f, dnn = 0.f, dee = 0.f;
  for (int k = 0; k < 64; ++k) {
    float e = emb[(size_t)node * 64 + k];
    float a = acc2[(size_t)node * 64 + k] - dg * e;
    dne += a * e; dnn += a * a; dee += e * e;
  }
  finish_score(node, N, dne, dnn, dee, dg, snum[node], outScores, keys);
}

__global__ void hist_kernel(const unsigned* __restrict__ keys,
                            unsigned* __restrict__ hist,
                            const unsigned* __restrict__ state, int N, int r) {
  int i = blockIdx.x * blockDim.x + threadIdx.x;
  if (i >= N) return;
  unsigned key = keys[i];
  bool match = (r == 0) || ((key >> ((4 - r) * 8)) == state[0]);
  if (match) {
    unsigned byte = (key >> ((3 - r) * 8)) & 255u;
    atomicAdd(&hist[r * 256 + byte], 1u);
  }
}

__global__ void select_kernel(const unsigned* __restrict__ hist,
                              unsigned* __restrict__ state, int r) {
  if (threadIdx.x != 0) return;
  unsigned k = state[1];
  unsigned prefix = state[0];
  const unsigned* h = hist + r * 256;
  unsigned cum = 0;
  for (int b = 255; b >= 0; --b) {
    unsigned c = h[b];
    if (cum + c >= k) {
      state[0] = (prefix << 8) | (unsigned)b;
      state[1] = k - cum;
      return;
    }
    cum += c;
  }
  state[0] = prefix << 8;  // degenerate fallback
  state[1] = (k > cum) ? (k - cum) : 1u;
}

__global__ void collect_kernel(const unsigned* __restrict__ keys,
                               unsigned* __restrict__ state,
                               unsigned* __restrict__ candG,
                               unsigned* __restrict__ candE, int N) {
  int i = blockIdx.x * blockDim.x + threadIdx.x;
  if (i >= N) return;
  unsigned T = state[0];
  unsigned key = keys[i];
  if (key > T) {
    unsigned p = atomicAdd(&state[2], 1u);
    if (p < 2048u) { candG[2 * p] = key; candG[2 * p + 1] = (unsigned)i; }
  } else if (key == T) {
    unsigned p = atomicAdd(&state[3], 1u);
    if (p < 2048u) { candE[2 * p] = key; candE[2 * p + 1] = (unsigned)i; }
  }
}

// single-block bitonic sort of 2048 composite keys: (key desc, idx asc)
__global__ void __launch_bounds__(1024) topk_sort_kernel(
    const unsigned* __restrict__ state, const unsigned* __restrict__ candG,
    const unsigned* __restrict__ candE, const int* __restrict__ nseeds,
    float* __restrict__ d_out, int N) {
  __shared__ unsigned long long smem[2048];
  const unsigned t = threadIdx.x;
  int kk = nseeds[0];
  if (kk < 1) kk = 1;
  if (kk > 2048) kk = 2048;
  const unsigned K = (unsigned)kk;
  unsigned cntG = state[2]; if (cntG > K) cntG = K;
  unsigned cntE = state[3]; if (cntE > 2048u - cntG) cntE = 2048u - cntG;

  for (unsigned j = t; j < 2048u; j += 1024u) {
    unsigned key, idx;
    if (j < cntG)             { key = candG[2 * j];          idx = candG[2 * j + 1]; }
    else if (j < cntG + cntE) { unsigned q = j - cntG;
                                key = candE[2 * q];          idx = candE[2 * q + 1]; }
    else                      { key = 0u;                    idx = 0xFFFFFFFFu; }
    smem[j] = ((unsigned long long)key << 32) | (unsigned long long)(0xFFFFFFFFu - idx);
  }
  for (unsigned ksz = 2; ksz <= 2048u; ksz <<= 1) {
    for (unsigned jsz = ksz >> 1; jsz > 0; jsz >>= 1) {
      __syncthreads();
      for (unsigned base = t; base < 2048u; base += 1024u) {
        unsigned partner = base ^ jsz;
        if (partner > base) {
          bool desc = ((base & ksz) == 0u);
          unsigned long long a = smem[base], b = smem[partner];
          if (desc ? (a < b) : (a > b)) { smem[base] = b; smem[partner] = a; }
        }
      }
    }
  }
  __syncthreads();
  int* seeds = (int*)(d_out + N);  // int32 seed indices, bit-stored after scores
  for (unsigned j = t; j < K; j += 1024u) {
    unsigned idx = 0xFFFFFFFFu - (unsigned)(smem[j] & 0xFFFFFFFFu);
    seeds[j] = (int)idx;
  }
}

// ------------------------------ launcher -----------------------------------

static inline int cdiv_ll(long long a, long long b) {
  return (int)((a + b - 1) / b);
}

extern "C" void kernel_launch(void* const* d_in, const int* in_sizes, int n_in,
                              void* d_out, int out_size, void* d_ws,
                              size_t ws_size, hipStream_t stream) {
  const float* emb = (const float*)d_in[0];
  const int* ei = (const int*)d_in[1];
  const int* nseeds = (const int*)d_in[3];
  float* out = (float*)d_out;

  const int N = in_sizes[0] / 64;
  const int E = in_sizes[1] / 2;

  // workspace layout (4-byte words)
  float* deg = (float*)d_ws;                    // N
  float* snum = deg + N;                        // N
  float* acc1 = snum + N;                       // N*64  (spmm1 -> first_emb)
  float* acc2 = acc1 + (size_t)N * 64;          // N*64  (spmm2)
  unsigned* keys = (unsigned*)(acc2 + (size_t)N * 64);  // N
  unsigned* hist = keys + N;                    // 4*256
  unsigned* state = hist + 1024;                // 8
  unsigned* candG = state + 8;                  // 2048*2
  unsigned* candE = candG + 4096;               // 2048*2
  const size_t wsWords = (size_t)N * 131 + 1024 + 8 + 4096 + 4096;

  zero_kernel<<<2048, 256, 0, stream>>>((unsigned*)d_ws, wsWords);
  init_state_kernel<<<1, 32, 0, stream>>>(state, nseeds);

  degree_kernel<<<cdiv_ll(E, 256), 256, 0, stream>>>(ei, deg, E);
  spmm_edge_kernel<<<cdiv_ll((long long)E * 32, 256), 256, 0, stream>>>(
      ei, emb, acc1, E);
  sub_kernel<<<cdiv_ll((long long)N * 64, 256), 256, 0, stream>>>(
      acc1, emb, (long long)N * 64);
  spmm_edge_kernel<<<cdiv_ll((long long)E * 32, 256), 256, 0, stream>>>(
      ei, acc1, acc2, E);
  spmm_num_kernel<<<cdiv_ll(E, 256), 256, 0, stream>>>(ei, deg, snum, E);

  const int ntiles = N >> 4;
  if (ntiles > 0)
    score_tile_kernel<<<cdiv_ll(ntiles, 8), 256, 0, stream>>>(
        emb, acc2, deg, snum, out, keys, N);
  const int tailStart = ntiles << 4;
  if (tailStart < N)
    score_tail_kernel<<<cdiv_ll(N - tailStart, 256), 256, 0, stream>>>(
        emb, acc2, deg, snum, out, keys, N, tailStart);

  for (int r = 0; r < 4; ++r) {
    hist_kernel<<<cdiv_ll(N, 256), 256, 0, stream>>>(keys, hist, state, N, r);
    select_kernel<<<1, 32, 0, stream>>>(hist, state, r);
  }
  collect_kernel<<<cdiv_ll(N, 256), 256, 0, stream>>>(keys, state, candG,
                                                      candE, N);
  topk_sort_kernel<<<1, 1024, 0, stream>>>(state, candG, candE, nseeds, out, N);
}